// SparseMaxPool_41085657154000
// MI455X (gfx1250) — compile-verified
//
#include <hip/hip_runtime.h>
#include <cstdint>

// out(i,j) = mask(i,j) ? max(x[i..j]) : 0   for each (b,d) row of 64 floats.
// mask: off=j-i in [0,15] always; off odd in [17,31] when i%2==0;
//       off≡3 (mod 4) in [35,63] when i%4==0.
//
// Per row: 7-level sparse table in LDS (padded stride 65 to kill bank
// conflicts), built from a row staged by gfx1250 async global->LDS DMA.
// Output (268 MB, write-once) streamed with non-temporal b128 stores.

typedef float v4f __attribute__((ext_vector_type(4)));

#define WAVES_PER_BLOCK 8
#define LVL_STRIDE 65                    // 64 + 1 pad: bank = (65k+i)%64 = (k+i)%64
#define WAVE_LDS   (7 * LVL_STRIDE)      // 455 floats per wave (levels k=0..6)

__global__ __launch_bounds__(256) void sparse_maxpool_kernel(
    const float* __restrict__ x, float* __restrict__ out, int nrows)
{
    __shared__ float sh[WAVES_PER_BLOCK * WAVE_LDS];

    const int lane = threadIdx.x & 31;
    const int wave = threadIdx.x >> 5;
    const int row  = blockIdx.x * WAVES_PER_BLOCK + wave;
    float* shw = sh + wave * WAVE_LDS;

    // ---- stage this row's 64 floats into LDS level 0 via async DMA ----
    // Each lane moves 8 bytes: 32 lanes x b64 = 256 B = whole row.
    if (row < nrows) {
        uint32_t lds_addr = (uint32_t)(uintptr_t)(shw + lane * 2);
        const float* gaddr = x + (size_t)row * 64 + lane * 2;
        asm volatile("global_load_async_to_lds_b64 %0, %1, off"
                     :: "v"(lds_addr), "v"(gaddr) : "memory");
    }
    asm volatile("s_wait_asynccnt 0" ::: "memory");

    // ---- build sparse table: ST[k][i] = max(x[i .. min(i+2^k-1, 63)]) ----
    for (int k = 1; k <= 6; ++k) {
        __syncthreads();
        if (row < nrows) {
            const int h = 1 << (k - 1);
            #pragma unroll
            for (int u = 0; u < 2; ++u) {
                int i  = lane + u * 32;
                int ip = i + h; ip = ip > 63 ? 63 : ip;
                shw[k * LVL_STRIDE + i] =
                    fmaxf(shw[(k - 1) * LVL_STRIDE + i],
                          shw[(k - 1) * LVL_STRIDE + ip]);
            }
        }
    }
    __syncthreads();

    if (row >= nrows) return;

    // ---- emit the 64x64 tile: 1024 float4 per row, 32 per lane ----
    float* orow = out + (size_t)row * 4096;
    for (int t = 0; t < 32; ++t) {
        const int idx = t * 32 + lane;       // consecutive lanes -> contiguous 16B
        const int i   = idx >> 4;
        const int j0  = (idx & 15) << 2;
        v4f v;
        #pragma unroll
        for (int c = 0; c < 4; ++c) {
            const int j   = j0 + c;
            const int off = j - i;
            bool m;
            if (off < 0)        m = false;
            else if (off <= 15) m = true;                               // incl. diagonal
            else if (off <= 31) m = (off & 1) && !(i & 1);              // odd 17..31
            else                m = (off >= 35) && ((off & 3) == 3) && !(i & 3);
            float val = 0.0f;
            if (m) {
                const int len = off + 1;                                // 1..64
                const int k   = 31 - __builtin_clz((unsigned)len);      // floor log2
                val = fmaxf(shw[k * LVL_STRIDE + i],
                            shw[k * LVL_STRIDE + j - (1 << k) + 1]);
            }
            v[c] = val;
        }
        // 268 MB write-once stream > 192 MB L2: keep it non-temporal.
        __builtin_nontemporal_store(v, (v4f*)(orow + (size_t)idx * 4));
    }
}

extern "C" void kernel_launch(void* const* d_in, const int* in_sizes, int n_in,
                              void* d_out, int out_size, void* d_ws, size_t ws_size,
                              hipStream_t stream) {
    (void)n_in; (void)out_size; (void)d_ws; (void)ws_size;
    const float* x = (const float*)d_in[0];
    float* out     = (float*)d_out;
    const int nrows  = in_sizes[0] / 64;                 // B*D = 16384
    const int blocks = (nrows + WAVES_PER_BLOCK - 1) / WAVES_PER_BLOCK;
    sparse_maxpool_kernel<<<blocks, 256, 0, stream>>>(x, out, nrows);
}